// HQNN_25486335935171
// MI455X (gfx1250) — compile-verified
//
#include <hip/hip_runtime.h>
#include <hip/hip_bf16.h>

typedef __attribute__((ext_vector_type(2))) float v2f;
typedef __attribute__((ext_vector_type(8))) float v8f;

// ---------------------------------------------------------------------------
// Kernel 1: one wave32 builds the 16x4 output table.
// Lane L simulates basis input m = L&15 (lanes 16..31 duplicate 0..15, which
// is exactly what the WMMA A-layout needs: low half supplies K=0,1, high
// half supplies K=2,3). Linear(4,4)+bias done with V_WMMA_F32_16X16X4_F32.
// ---------------------------------------------------------------------------
__global__ __launch_bounds__(32) void HQNN_table_kernel(
    const float* __restrict__ weights,  // [4][4][3] (layer, wire, {phi,theta,omega})
    const float* __restrict__ fc_w,     // [4][4] row-major (out_features, in_features)
    const float* __restrict__ fc_b,     // [4]
    float* __restrict__ table)          // [16][4]
{
    const int lane = threadIdx.x;
    const int m    = lane & 15;

    // Post-encoding state is exactly |m> (RZ(pi)RX(pi) maps |0>->+|1>).
    float sr[16], si[16];
#pragma unroll
    for (int i = 0; i < 16; ++i) { sr[i] = 0.0f; si[i] = 0.0f; }
    sr[m] = 1.0f;

    // 4 variational layers: CNOT ring (0,1)(1,2)(2,3)(3,0) then Rot per wire.
#pragma unroll
    for (int l = 0; l < 4; ++l) {
        const int cw[4] = {0, 1, 2, 3};
        const int tw[4] = {1, 2, 3, 0};
#pragma unroll
        for (int g = 0; g < 4; ++g) {
            const int pc = 3 - cw[g];
            const int pt = 3 - tw[g];
#pragma unroll
            for (int idx = 0; idx < 16; ++idx) {
                if ((((idx >> pc) & 1) == 1) && (((idx >> pt) & 1) == 0)) {
                    const int j = idx | (1 << pt);
                    float tr = sr[idx], ti = si[idx];
                    sr[idx] = sr[j]; si[idx] = si[j];
                    sr[j] = tr;      si[j] = ti;
                }
            }
        }
#pragma unroll
        for (int w = 0; w < 4; ++w) {
            const float phi   = weights[(l * 4 + w) * 3 + 0];
            const float theta = weights[(l * 4 + w) * 3 + 1];
            const float omega = weights[(l * 4 + w) * 3 + 2];
            const float ch = cosf(0.5f * theta);
            const float sh = sinf(0.5f * theta);
            float csp, ssp, cdm, sdm;
            sincosf(0.5f * (phi + omega), &ssp, &csp);
            sincosf(0.5f * (phi - omega), &sdm, &cdm);
            // U = [[a, b],[c, d]] (complex)
            const float ar =  ch * csp, ai = -ch * ssp;
            const float br = -sh * cdm, bi = -sh * sdm;
            const float cr =  sh * cdm, ci = -sh * sdm;
            const float dr =  ch * csp, di =  ch * ssp;
            const int p = 3 - w, mask = 1 << p;
#pragma unroll
            for (int idx = 0; idx < 16; ++idx) {
                if ((idx & mask) == 0) {
                    const int j = idx | mask;
                    const float r0 = sr[idx], i0 = si[idx];
                    const float r1 = sr[j],   i1 = si[j];
                    sr[idx] = ar * r0 - ai * i0 + br * r1 - bi * i1;
                    si[idx] = ar * i0 + ai * r0 + br * i1 + bi * r1;
                    sr[j]   = cr * r0 - ci * i0 + dr * r1 - di * i1;
                    si[j]   = cr * i0 + ci * r0 + dr * i1 + di * r1;
                }
            }
        }
    }

    // <Z_w> expectations per lane's basis input.
    float ev[4] = {0.0f, 0.0f, 0.0f, 0.0f};
#pragma unroll
    for (int idx = 0; idx < 16; ++idx) {
        const float p = sr[idx] * sr[idx] + si[idx] * si[idx];
#pragma unroll
        for (int w = 0; w < 4; ++w)
            ev[w] += (((idx >> (3 - w)) & 1) ? -p : p);
    }

    // ---- Linear(4,4) + bias via V_WMMA_F32_16X16X4_F32: D = A(16x4) B(4x16) + C
    const bool hi  = (lane & 16) != 0;  // high half supplies K=2,3
    const int  col = lane & 15;         // N index for B/C/D

    v2f A;                               // A[m][k]: k = hi?{2,3}:{0,1}
    A.x = hi ? ev[2] : ev[0];
    A.y = hi ? ev[3] : ev[1];

    v2f Bv;                              // B[k][n] = fc_w[n][k], zero-padded n>=4
    if (col < 4) {
        Bv.x = fc_w[col * 4 + (hi ? 2 : 0)];
        Bv.y = fc_w[col * 4 + (hi ? 3 : 1)];
    } else {
        Bv.x = 0.0f; Bv.y = 0.0f;
    }

    const float cbias = (col < 4) ? fc_b[col] : 0.0f;  // C[m][n] = fc_b[n]
    v8f C;
#pragma unroll
    for (int r = 0; r < 8; ++r) C[r] = cbias;

    v8f D = __builtin_amdgcn_wmma_f32_16x16x4_f32(
        /*neg_a=*/false, A, /*neg_b=*/false, Bv,
        /*c_mod=*/(short)0, C, /*reuse_a=*/false, /*reuse_b=*/false);

    if (col < 4) {
        const int base = hi ? 8 : 0;     // D VGPR r -> row r (+8 for high half)
#pragma unroll
        for (int r = 0; r < 8; ++r) {
            float v = D[r];
            v = (v >= 0.0f) ? v : 0.01f * v;   // leaky_relu
            table[(base + r) * 4 + col] = v;
        }
    }
}

// ---------------------------------------------------------------------------
// Kernel 2: bandwidth-bound gather. 4 samples/thread: int4 load of x,
// 16-entry LDS table, 4 contiguous float4 stores (64B/thread).
// ---------------------------------------------------------------------------
__global__ __launch_bounds__(256) void HQNN_gather_kernel(
    const int* __restrict__ x,
    const float* __restrict__ table,   // [16][4]
    float* __restrict__ out,           // [B][4]
    int B)
{
    __shared__ float4 t[16];
    if (threadIdx.x < 16)
        t[threadIdx.x] = reinterpret_cast<const float4*>(table)[threadIdx.x];
    __syncthreads();

    const int q  = blockIdx.x * blockDim.x + threadIdx.x;
    const int i0 = q * 4;
    if (i0 + 3 < B) {
        const int4 xs = reinterpret_cast<const int4*>(x)[q];
        float4* o = reinterpret_cast<float4*>(out) + i0;
        o[0] = t[xs.x & 15];
        o[1] = t[xs.y & 15];
        o[2] = t[xs.z & 15];
        o[3] = t[xs.w & 15];
    } else if (i0 < B) {
        for (int i = i0; i < B; ++i)
            reinterpret_cast<float4*>(out)[i] = t[x[i] & 15];
    }
}

extern "C" void kernel_launch(void* const* d_in, const int* in_sizes, int n_in,
                              void* d_out, int out_size, void* d_ws, size_t ws_size,
                              hipStream_t stream) {
    const int*   x       = (const int*)d_in[0];
    const float* weights = (const float*)d_in[1];
    const float* fc_w    = (const float*)d_in[2];
    const float* fc_b    = (const float*)d_in[3];
    float* out   = (float*)d_out;
    float* table = (float*)d_ws;           // 64 floats of scratch
    const int B  = in_sizes[0];

    HQNN_table_kernel<<<1, 32, 0, stream>>>(weights, fc_w, fc_b, table);

    const int nq     = (B + 3) / 4;
    const int blocks = (nq + 255) / 256;
    HQNN_gather_kernel<<<blocks, 256, 0, stream>>>(x, table, out, B);
}